// FluxMapper_12859132084977
// MI455X (gfx1250) — compile-verified
//
#include <hip/hip_runtime.h>

// FluxMapper: flux[b,e] = 0.5*(node[b,src[e],:] + node[b,dst[e],:]) . edge_vec[e,:]
// B=8, D=3. Memory-bound streaming gather; async global->LDS double-buffer pipeline.

#define TILE  256
#define BATCH 8

// Async global->LDS builtins are gfx1250-only; probe in the device pass.
#if defined(__AMDGCN__) && defined(__has_builtin)
#  if __has_builtin(__builtin_amdgcn_global_load_async_to_lds_b32) && \
      __has_builtin(__builtin_amdgcn_s_wait_asynccnt)
#    define USE_ASYNC 1
#  endif
#endif

typedef __attribute__((address_space(1))) int gint;  // global (AS1)
typedef __attribute__((address_space(3))) int lint;  // LDS    (AS3)

__device__ __forceinline__ void flux_compute(
    const float* __restrict__ node, float* __restrict__ out,
    long long e, long long E, long long N3,
    int s, int d, float ex, float ey, float ez)
{
    const float* ps = node + 3ll * (long long)s;
    const float* pd = node + 3ll * (long long)d;
#pragma unroll
    for (int b = 0; b < BATCH; ++b) {
        const float* a = ps + (long long)b * N3;   // L2-resident gathers
        const float* c = pd + (long long)b * N3;
        float f = (a[0] + c[0]) * ex + (a[1] + c[1]) * ey + (a[2] + c[2]) * ez;
        out[(long long)b * E + e] = f;             // coalesced per batch
    }
}

__global__ __launch_bounds__(TILE)
void flux_kernel(const float* __restrict__ node, const float* __restrict__ ev,
                 const int* __restrict__ src, const int* __restrict__ dst,
                 float* __restrict__ out, long long E, long long N3, int ntiles)
{
    const int t = threadIdx.x;

#if USE_ASYNC
    __shared__ int   s_src[2][TILE];
    __shared__ int   s_dst[2][TILE];
    __shared__ float s_ev[2][3 * TILE];

    const int       stride = gridDim.x;
    const long long ftot   = 3ll * E;

    // Stage one 256-edge tile into LDS buffer `buf`: 5 async B32 loads/thread.
    // Addresses are clamped (not predicated) so every wave always issues
    // exactly 5 async ops -> s_wait_asynccnt math stays uniform.
    auto stage = [&](int tile, int buf) {
        long long e  = (long long)tile * TILE + t;
        long long ge = (e < E) ? e : (E - 1);
        __builtin_amdgcn_global_load_async_to_lds_b32(
            (gint*)(src + ge), (lint*)&s_src[buf][t], 0, 0);
        __builtin_amdgcn_global_load_async_to_lds_b32(
            (gint*)(dst + ge), (lint*)&s_dst[buf][t], 0, 0);
        long long fb = (long long)tile * (3 * TILE);
#pragma unroll
        for (int k = 0; k < 3; ++k) {
            long long fi = fb + t + k * TILE;      // coalesced float stream
            if (fi >= ftot) fi = ftot - 1;
            __builtin_amdgcn_global_load_async_to_lds_b32(
                (gint*)(ev + fi), (lint*)&s_ev[buf][t + k * TILE], 0, 0);
        }
    };

    int tile = blockIdx.x;
    int buf  = 0;
    stage(tile, 0);

    for (; tile < ntiles; tile += stride) {
        // Prefetch two tiles ahead into L2 (global_prefetch_b8).
        long long pfTile = (long long)tile + 2ll * stride;
        if (pfTile < ntiles) {
            long long pe = pfTile * TILE + t;
            if (pe < E) {
                __builtin_prefetch(src + pe, 0, 1);
                __builtin_prefetch(dst + pe, 0, 1);
                __builtin_prefetch(ev + 3ll * pe, 0, 1);
            }
        }

        // Kick off the next tile into the other buffer, then retire the 5
        // older ops (this buffer). Async loads complete in order per wave.
        int nxt = tile + stride;
        if (nxt >= ntiles) nxt = ntiles - 1;       // redundant clamp at tail
        stage(nxt, buf ^ 1);
        __builtin_amdgcn_s_wait_asynccnt(5);
        __syncthreads();                            // all waves' stages landed

        long long e = (long long)tile * TILE + t;
        if (e < E) {
            int   s  = s_src[buf][t];
            int   d  = s_dst[buf][t];
            float ex = 0.5f * s_ev[buf][3 * t + 0];
            float ey = 0.5f * s_ev[buf][3 * t + 1];
            float ez = 0.5f * s_ev[buf][3 * t + 2];
            flux_compute(node, out, e, E, N3, s, d, ex, ey, ez);
        }
        __syncthreads();                            // buffer reusable
        buf ^= 1;
    }
#else
    // Fallback: direct grid-stride streaming (no async builtins available).
    for (long long e = (long long)blockIdx.x * TILE + t; e < E;
         e += (long long)gridDim.x * TILE) {
        __builtin_prefetch(ev + 3ll * e + 3ll * (long long)gridDim.x * TILE, 0, 1);
        int   s  = src[e];
        int   d  = dst[e];
        float ex = 0.5f * ev[3 * e + 0];
        float ey = 0.5f * ev[3 * e + 1];
        float ez = 0.5f * ev[3 * e + 2];
        flux_compute(node, out, e, E, N3, s, d, ex, ey, ez);
    }
#endif
}

extern "C" void kernel_launch(void* const* d_in, const int* in_sizes, int n_in,
                              void* d_out, int out_size, void* d_ws, size_t ws_size,
                              hipStream_t stream) {
    (void)n_in; (void)out_size; (void)d_ws; (void)ws_size;
    const float* node = (const float*)d_in[0];   // [8, N, 3] f32
    const float* ev   = (const float*)d_in[1];   // [E, 3]    f32
    const int*   src  = (const int*)d_in[2];     // [E]       i32
    const int*   dst  = (const int*)d_in[3];     // [E]       i32
    float*       out  = (float*)d_out;           // [8, E]    f32

    long long E  = (long long)in_sizes[2];
    long long N3 = (long long)in_sizes[0] / BATCH;   // = 3*N
    int ntiles   = (int)((E + TILE - 1) / TILE);
    int grid     = ntiles < 2048 ? ntiles : 2048;    // persistent pipeline blocks

    flux_kernel<<<grid, TILE, 0, stream>>>(node, ev, src, dst, out, E, N3, ntiles);
}